// GINLayer_79688823210541
// MI455X (gfx1250) — compile-verified
//
#include <hip/hip_runtime.h>

typedef __attribute__((ext_vector_type(2))) float v2f;
typedef __attribute__((ext_vector_type(8))) float v8f;

#define DIM 256
#define LDS_PITCH 260   // pad: (4*m + k) % 64 distinct across lanes -> conflict-free ds_load_b64
#define MROWS 32        // rows per block (2 M-tiles)

// C[nrows,256] = act( (A [+ A2]) @ W + bias )
// grid.x = ceil(nrows/32); block = 256 threads (8 waves).
// Wave w computes a 2x2 grid of 16x16 tiles: rows {0-15,16-31} of the strip,
// columns {32w..32w+15, 32w+16..32w+31}. B fragments reused across both M-tiles,
// A fragments reused across both N-tiles -> 2 ds_b64 + 4 b32 loads per 4 WMMAs.
template<bool RELU, bool ADD2>
__global__ __launch_bounds__(256) void gemm_mlp_kernel(
    const float* __restrict__ A, const float* __restrict__ A2,
    const float* __restrict__ W, const float* __restrict__ bias,
    float* __restrict__ C, int nrows)
{
    __shared__ float sA[MROWS * LDS_PITCH];

    const int t = threadIdx.x;
    const int rowBase = blockIdx.x * MROWS;

    // Cooperative stage of the 32x256 A strip (optionally + A2) into LDS.
    for (int i = t * 4; i < MROWS * DIM; i += 256 * 4) {
        const int r = i / DIM;
        const int c = i % DIM;
        const int gr = rowBase + r;
        float4 v = make_float4(0.f, 0.f, 0.f, 0.f);
        if (gr < nrows) {
            v = *(const float4*)&A[(size_t)gr * DIM + c];
            if (ADD2) {
                const float4 u = *(const float4*)&A2[(size_t)gr * DIM + c];
                v.x += u.x; v.y += u.y; v.z += u.z; v.w += u.w;
            }
        }
        *(float4*)&sA[r * LDS_PITCH + c] = v;   // (r*260+c)*4 is 16B aligned
    }
    __syncthreads();

    const int lane = t & 31;
    const int wave = t >> 5;
    const int m    = lane & 15;           // A row within tile / B,C column within tile
    const int koff = (lane >> 4) << 1;    // upper half-wave holds K+2,K+3 (A rows / B rows)
    const int n0   = wave * 32 + m;       // first output column for this lane
    const int n1   = n0 + 16;             // second N-tile's column

    v8f acc00 = {};   // rows 0-15,  cols n0 tile
    v8f acc01 = {};   // rows 0-15,  cols n1 tile
    v8f acc10 = {};   // rows 16-31, cols n0 tile
    v8f acc11 = {};   // rows 16-31, cols n1 tile

    const float* sA0 = &sA[m * LDS_PITCH];
    const float* sA1 = &sA[(m + 16) * LDS_PITCH];

    for (int k = 0; k < DIM; k += 4) {
        // A fragments: 16x4 f32 per M-tile, lane -> A[m][k+koff..+1]  (ds_load_b64 each)
        const v2f a0 = *(const v2f*)&sA0[k + koff];
        const v2f a1 = *(const v2f*)&sA1[k + koff];

        // B fragments: 4x16 f32 per N-tile, lane -> W[k+koff][n], W[k+koff+1][n]
        const int kb = k + koff;
        v2f b0, b1;
        b0.x = W[(size_t)kb * DIM + n0];
        b0.y = W[(size_t)(kb + 1) * DIM + n0];
        b1.x = W[(size_t)kb * DIM + n1];
        b1.y = W[(size_t)(kb + 1) * DIM + n1];

        acc00 = __builtin_amdgcn_wmma_f32_16x16x4_f32(false, a0, false, b0,
                                                      (short)0, acc00, false, false);
        acc01 = __builtin_amdgcn_wmma_f32_16x16x4_f32(false, a0, false, b1,
                                                      (short)0, acc01, false, false);
        acc10 = __builtin_amdgcn_wmma_f32_16x16x4_f32(false, a1, false, b0,
                                                      (short)0, acc10, false, false);
        acc11 = __builtin_amdgcn_wmma_f32_16x16x4_f32(false, a1, false, b1,
                                                      (short)0, acc11, false, false);
    }

    // Epilogue: bias, optional ReLU, store 2x2 tiles.
    const float bv0 = bias[n0];
    const float bv1 = bias[n1];
    const int rbase = rowBase + ((lane >> 4) << 3);  // lanes 16-31 hold rows M+8
    #pragma unroll
    for (int v = 0; v < 8; ++v) {
        const int r0 = rbase + v;        // M-tile 0
        const int r1 = r0 + 16;          // M-tile 1
        if (r0 < nrows) {
            float c0 = acc00[v] + bv0;
            float c1 = acc01[v] + bv1;
            if (RELU) { c0 = fmaxf(c0, 0.f); c1 = fmaxf(c1, 0.f); }
            C[(size_t)r0 * DIM + n0] = c0;
            C[(size_t)r0 * DIM + n1] = c1;
        }
        if (r1 < nrows) {
            float c2 = acc10[v] + bv0;
            float c3 = acc11[v] + bv1;
            if (RELU) { c2 = fmaxf(c2, 0.f); c3 = fmaxf(c3, 0.f); }
            C[(size_t)r1 * DIM + n0] = c2;
            C[(size_t)r1 * DIM + n1] = c3;
        }
    }
}

// agg[row[e]][:] += M[col[e]][:]  -- 64 lanes per edge, float4 per lane, 4 edges per block
__global__ __launch_bounds__(256) void scatter_add_kernel(
    const float* __restrict__ Msg, const long long* __restrict__ erow,
    const long long* __restrict__ ecol, float* __restrict__ agg, int E)
{
    const int t = threadIdx.x;
    const int e = blockIdx.x * 4 + (t >> 6);
    if (e >= E) return;
    const int l4 = (t & 63) * 4;
    const int row = (int)erow[e];
    const int col = (int)ecol[e];
    const float4 v = *(const float4*)&Msg[(size_t)col * DIM + l4];
    float* dst = &agg[(size_t)row * DIM + l4];
    atomicAdd(dst + 0, v.x);
    atomicAdd(dst + 1, v.y);
    atomicAdd(dst + 2, v.z);
    atomicAdd(dst + 3, v.w);
}

extern "C" void kernel_launch(void* const* d_in, const int* in_sizes, int n_in,
                              void* d_out, int out_size, void* d_ws, size_t ws_size,
                              hipStream_t stream) {
    const float*     x    = (const float*)d_in[0];
    const long long* edge = (const long long*)d_in[1];  // [2, E] int64
    const float*     w1   = (const float*)d_in[2];
    const float*     b1   = (const float*)d_in[3];
    const float*     w2   = (const float*)d_in[4];
    const float*     b2   = (const float*)d_in[5];
    float*           out  = (float*)d_out;

    const int N = in_sizes[0] / DIM;  // 10000
    const int E = in_sizes[1] / 2;    // 320000

    float* buf0 = (float*)d_ws;                    // H1, then agg
    float* buf1 = buf0 + (size_t)N * DIM;          // M (per-node messages), then T

    const dim3 blk(256);
    const int gM = (N + MROWS - 1) / MROWS;

    // H1 = relu(x @ w1 + b1)
    gemm_mlp_kernel<true,  false><<<gM, blk, 0, stream>>>(x, nullptr, w1, b1, buf0, N);
    // M = H1 @ w2 + b2   (per-node MLP output; MLP(x[col]) == M[col])
    gemm_mlp_kernel<false, false><<<gM, blk, 0, stream>>>(buf0, nullptr, w2, b2, buf1, N);
    // agg = 0
    hipMemsetAsync(buf0, 0, (size_t)N * DIM * sizeof(float), stream);
    // agg[row[e]] += M[col[e]]
    scatter_add_kernel<<<(E + 3) / 4, blk, 0, stream>>>(buf1, edge, edge + E, buf0, E);
    // T = relu((x + agg) @ w1 + b1)   (residual add fused into LDS stage)
    gemm_mlp_kernel<true,  true ><<<gM, blk, 0, stream>>>(x, buf0, w1, b1, buf1, N);
    // out = T @ w2 + b2
    gemm_mlp_kernel<false, false><<<gM, blk, 0, stream>>>(buf1, nullptr, w2, b2, out, N);
}